// Net_20220706030448
// MI455X (gfx1250) — compile-verified
//
#include <hip/hip_runtime.h>

#define T_STEPS 1000
#define BATCH   256
#define IN_DIM  128
#define HID     512
#define OUT_DIM 32
#define ALPHA_F 0.2f
#define OMA_F   0.8f

typedef float v2f __attribute__((ext_vector_type(2)));
typedef float v4f __attribute__((ext_vector_type(4)));
typedef float v8f __attribute__((ext_vector_type(8)));

__device__ __forceinline__ v8f wmma_f32(v2f a, v2f b, v8f c) {
    // D = A(16x4) * B(4x16) + C(16x16), full fp32
    return __builtin_amdgcn_wmma_f32_16x16x4_f32(
        /*neg_a=*/false, a, /*neg_b=*/false, b,
        /*c_mod=*/(short)0, c, /*reuse_a=*/false, /*reuse_b=*/false);
}

// ---------------------------------------------------------------------------
// Weight prep: FRAGMENT-PACKED B matrices covering TWO k-quads per 16B.
// For B of shape [K,N] and k = 8*q2 + 4*qlo + 2*hi + j (j in {0,1}):
//   bpack[ (q2*N + n)*8 + hi*4 + qlo*2 + j ] = B[k][n]
// A lane (hi = lane>>4) fetches ONE aligned float4 = both WMMA B fragments
// (quads 2*q2 and 2*q2+1):  ((v4f*)bpack)[(q2*N + n)*2 + hi]
// A 4x16 B tile-pair is a contiguous 512-byte block (fully coalesced b128s).
// ---------------------------------------------------------------------------

__device__ __forceinline__ size_t pack_off(int k, int n, int N) {
    return ((size_t)(k >> 3) * N + n) * 8 + ((k >> 1) & 1) * 4 + ((k >> 2) & 1) * 2 + (k & 1);
}

// B(k=h2, n=h) = relu(w_h[h,h2]) * dale[h2] * sparse[h,h2]
__global__ void prep_weff(const float* __restrict__ wh, const float* __restrict__ dale,
                          const float* __restrict__ sp, float* __restrict__ weffP) {
    int idx = blockIdx.x * 256 + threadIdx.x;
    if (idx >= HID * HID) return;
    int h  = idx / HID;   // postsynaptic -> N index
    int h2 = idx % HID;   // presynaptic  -> K index
    float v = wh[idx];
    v = v > 0.f ? v : 0.f;
    v *= dale[h2] * sp[idx];
    weffP[pack_off(h2, h, HID)] = v;
}

// B(k=i, n=h) = w_in[h, i]
__global__ void prep_win(const float* __restrict__ win, float* __restrict__ winP) {
    int idx = blockIdx.x * 256 + threadIdx.x;
    if (idx >= HID * IN_DIM) return;
    int h = idx / IN_DIM;
    int i = idx % IN_DIM;
    winP[pack_off(i, h, HID)] = win[idx];
}

// B(k=h, n=o) = fc_w[o, h]
__global__ void prep_fc(const float* __restrict__ fcw, float* __restrict__ fcP) {
    int idx = blockIdx.x * 256 + threadIdx.x;
    if (idx >= OUT_DIM * HID) return;
    int o = idx / HID;
    int h = idx % HID;
    fcP[pack_off(h, o, OUT_DIM)] = fcw[idx];
}

// ---- fused persistent scan kernel: one WG per 16-row batch tile ----

#define LDO (HID + 4)     // padded LDS stride: b64 column reads hit 64 distinct banks
#define LDX (IN_DIM + 4)  // 132

__global__ __launch_bounds__(256) void rnn_scan(
    const float* __restrict__ x,     // [T,B,I]
    const float* __restrict__ b_in,  // [H]
    const float* __restrict__ b_h,   // [H]
    const float* __restrict__ fc_b,  // [O]
    const float* __restrict__ weffP, // packed [H/8][H][8]
    const float* __restrict__ winP,  // packed [I/8][H][8]
    const float* __restrict__ fcP,   // packed [H/8][O][8]
    float* __restrict__ out,         // [T,B,O]
    float* __restrict__ act)         // [T,B,H]
{
    __shared__ float s_out[16 * LDO];  // previous-step ReLU output, 16 rows x H
    __shared__ float s_x[16 * LDX];    // current x tile, 16 rows x I

    const int tid  = threadIdx.x;
    const int wv   = tid >> 5;         // 8 waves
    const int lane = tid & 31;
    const int m    = lane & 15;        // A-row / B-col / C-col index
    const int hi   = lane >> 4;
    const int b0   = blockIdx.x * 16;  // batch tile origin
    const int n_base = wv * 64;        // each wave owns 4 N-tiles (64 hidden cols)

    const v4f* __restrict__ weffV = (const v4f*)weffP;
    const v4f* __restrict__ winV  = (const v4f*)winP;
    const v4f* __restrict__ fcV   = (const v4f*)fcP;

    for (int idx = tid; idx < 16 * LDO; idx += 256) s_out[idx] = 0.f;

    // per-lane column biases (column fixed per lane in C/D layout)
    float biasH[4];
#pragma unroll
    for (int nt = 0; nt < 4; ++nt) {
        int h = n_base + nt * 16 + m;
        biasH[nt] = b_in[h] + b_h[h];
    }

    v8f state[4];
#pragma unroll
    for (int nt = 0; nt < 4; ++nt) state[nt] = (v8f){};

    __syncthreads();

    for (int t = 0; t < T_STEPS; ++t) {
        // cooperative coalesced load of x[t, b0:b0+16, :]
        for (int idx = tid; idx < 16 * IN_DIM; idx += 256) {
            int r = idx >> 7;          // IN_DIM == 128
            int i = idx & 127;
            s_x[r * LDX + i] = x[((size_t)(t * BATCH + b0 + r)) * IN_DIM + i];
        }
        __syncthreads();

        v8f acc[4];
#pragma unroll
        for (int nt = 0; nt < 4; ++nt) acc[nt] = (v8f){};

        // 1) input drive: A = x tile (16xI from LDS), B = packed w_in, K = 128
        for (int q2 = 0; q2 < IN_DIM / 8; ++q2) {
            v2f a0 = *(const v2f*)&s_x[m * LDX + 8 * q2 + 2 * hi];
            v2f a1 = *(const v2f*)&s_x[m * LDX + 8 * q2 + 4 + 2 * hi];
#pragma unroll
            for (int nt = 0; nt < 4; ++nt) {
                int n = n_base + nt * 16 + m;
                v4f b4 = winV[((size_t)q2 * HID + n) * 2 + hi];
                v2f blo; blo.x = b4.x; blo.y = b4.y;
                v2f bhi; bhi.x = b4.z; bhi.y = b4.w;
                acc[nt] = wmma_f32(a0, blo, acc[nt]);
                acc[nt] = wmma_f32(a1, bhi, acc[nt]);
            }
        }
        // 2) recurrence: A = out_{t-1} (16xH in LDS), B = packed w_eff, K = 512
        for (int q2 = 0; q2 < HID / 8; ++q2) {
            v2f a0 = *(const v2f*)&s_out[m * LDO + 8 * q2 + 2 * hi];
            v2f a1 = *(const v2f*)&s_out[m * LDO + 8 * q2 + 4 + 2 * hi];
#pragma unroll
            for (int nt = 0; nt < 4; ++nt) {
                int n = n_base + nt * 16 + m;
                v4f b4 = weffV[((size_t)q2 * HID + n) * 2 + hi];
                v2f blo; blo.x = b4.x; blo.y = b4.y;
                v2f bhi; bhi.x = b4.z; bhi.y = b4.w;
                acc[nt] = wmma_f32(a0, blo, acc[nt]);
                acc[nt] = wmma_f32(a1, bhi, acc[nt]);
            }
        }
        __syncthreads();  // everyone done reading old s_out

        // 3) leaky integration + ReLU; publish new output
#pragma unroll
        for (int nt = 0; nt < 4; ++nt) {
            int n = n_base + nt * 16 + m;
#pragma unroll
            for (int r = 0; r < 8; ++r) {
                float total = acc[nt][r] + biasH[nt];
                float s = state[nt][r] * OMA_F + total * ALPHA_F;
                state[nt][r] = s;
                float o = s > 0.f ? s : 0.f;
                int row = r + 8 * hi;
                s_out[row * LDO + n] = o;
                act[((size_t)(t * BATCH + b0 + row)) * HID + n] = o;
            }
        }
        __syncthreads();  // new s_out visible

        // 4) FC head: O = 32 -> 2 N-tiles on waves 0..1, TWO independent
        //    accumulation chains (K split in half) to shorten the serial chain.
        if (wv < 2) {
            v8f oaccA = (v8f){};
            v8f oaccB = (v8f){};
            int n = wv * 16 + m;
            const int QH = HID / 16;   // 32 octets per half
            for (int q2 = 0; q2 < QH; ++q2) {
                int qa = q2, qb = q2 + QH;
                v2f a0 = *(const v2f*)&s_out[m * LDO + 8 * qa + 2 * hi];
                v2f a1 = *(const v2f*)&s_out[m * LDO + 8 * qa + 4 + 2 * hi];
                v2f c0 = *(const v2f*)&s_out[m * LDO + 8 * qb + 2 * hi];
                v2f c1 = *(const v2f*)&s_out[m * LDO + 8 * qb + 4 + 2 * hi];
                v4f b4 = fcV[((size_t)qa * OUT_DIM + n) * 2 + hi];
                v4f d4 = fcV[((size_t)qb * OUT_DIM + n) * 2 + hi];
                v2f blo; blo.x = b4.x; blo.y = b4.y;
                v2f bhi; bhi.x = b4.z; bhi.y = b4.w;
                v2f dlo; dlo.x = d4.x; dlo.y = d4.y;
                v2f dhi; dhi.x = d4.z; dhi.y = d4.w;
                oaccA = wmma_f32(a0, blo, oaccA);
                oaccB = wmma_f32(c0, dlo, oaccB);
                oaccA = wmma_f32(a1, bhi, oaccA);
                oaccB = wmma_f32(c1, dhi, oaccB);
            }
            v8f oacc = oaccA + oaccB;
            float fb = fc_b[n];
#pragma unroll
            for (int r = 0; r < 8; ++r) {
                int row = r + 8 * hi;
                out[((size_t)(t * BATCH + b0 + row)) * OUT_DIM + n] = oacc[r] + fb;
            }
        }
        // FC(t) reads of s_out finish before any wave passes the t+1 barriers
        // that precede the next s_out write, so no extra barrier needed here.
    }
}

extern "C" void kernel_launch(void* const* d_in, const int* in_sizes, int n_in,
                              void* d_out, int out_size, void* d_ws, size_t ws_size,
                              hipStream_t stream) {
    const float* x      = (const float*)d_in[0];
    const float* w_in   = (const float*)d_in[1];
    const float* b_in   = (const float*)d_in[2];
    const float* w_h    = (const float*)d_in[3];
    const float* b_h    = (const float*)d_in[4];
    const float* dale   = (const float*)d_in[5];
    const float* sparse = (const float*)d_in[6];
    const float* fc_w   = (const float*)d_in[7];
    const float* fc_b   = (const float*)d_in[8];

    float* out = (float*)d_out;                                   // [T,B,O]
    float* act = out + (size_t)T_STEPS * BATCH * OUT_DIM;         // [T,B,H]

    float* weffP = (float*)d_ws;                    // H*H floats  (1 MB)
    float* winP  = weffP + (size_t)HID * HID;       // I*H floats  (256 KB)
    float* fcP   = winP + (size_t)IN_DIM * HID;     // H*O floats  (64 KB)

    prep_weff<<<(HID * HID + 255) / 256, 256, 0, stream>>>(w_h, dale, sparse, weffP);
    prep_win<<<(HID * IN_DIM + 255) / 256, 256, 0, stream>>>(w_in, winP);
    prep_fc<<<(OUT_DIM * HID + 255) / 256, 256, 0, stream>>>(fc_w, fcP);

    rnn_scan<<<BATCH / 16, 256, 0, stream>>>(x, b_in, b_h, fc_b,
                                             weffP, winP, fcP, out, act);
}